// GNNV2_18021682774979
// MI455X (gfx1250) — compile-verified
//
#include <hip/hip_runtime.h>

// Problem constants (from reference): feat_pos (16, 320, 32, 32) f32
#define BATCH 16
#define CT    320
#define NPTS  1024   // 32*32
#define DIMF  256    // feature channels
#define POSD  64     // position channels (320-256)
#define KSEL  32     // top-k

typedef __attribute__((ext_vector_type(2))) float v2f;
typedef __attribute__((ext_vector_type(8))) float v8f;

// ---------------------------------------------------------------------------
// Kernel 1: transpose feat (b,256,1024) -> featT (b,1024,256) so that a
// gathered feature row is a contiguous 1KB chunk (coalesced 128B loads/wave).
// ---------------------------------------------------------------------------
__global__ __launch_bounds__(256) void feat_transpose(
    const float* __restrict__ in, float* __restrict__ featT) {
  __shared__ float tile[32][33];
  const int b  = blockIdx.z;
  const int n0 = blockIdx.x * 32;
  const int c0 = blockIdx.y * 32;
  const int tx = threadIdx.x;        // 0..31 (fast)
  const int ty = threadIdx.y;        // 0..7
#pragma unroll
  for (int i = 0; i < 4; ++i) {
    int c = c0 + ty + i * 8;
    tile[ty + i * 8][tx] = in[(size_t)b * CT * NPTS + (size_t)c * NPTS + n0 + tx];
  }
  __syncthreads();
#pragma unroll
  for (int i = 0; i < 4; ++i) {
    int n = n0 + ty + i * 8;
    featT[(size_t)b * NPTS * DIMF + (size_t)n * DIMF + c0 + tx] = tile[tx][ty + i * 8];
  }
}

// ---------------------------------------------------------------------------
// Kernel 2: normalize pos channels into posn (b,1024,64) row-major: each
// point's 64-d unit vector is a contiguous 256B row (feeds WMMA frag loads).
// ---------------------------------------------------------------------------
__global__ __launch_bounds__(256) void pos_normalize(
    const float* __restrict__ in, float* __restrict__ posn) {
  const int b = blockIdx.y;
  const int n = blockIdx.x * 256 + threadIdx.x;
  const float* src = in + (size_t)b * CT * NPTS + (size_t)DIMF * NPTS + n;
  float v[POSD];
  float ss = 0.0f;
#pragma unroll
  for (int k = 0; k < POSD; ++k) {
    v[k] = src[(size_t)k * NPTS];   // coalesced across threads (n fastest)
    ss += v[k] * v[k];
  }
  float inv = 1.0f / fmaxf(__builtin_sqrtf(ss), 1e-12f);
  float* dst = posn + ((size_t)b * NPTS + n) * POSD;
#pragma unroll
  for (int k = 0; k < POSD; ++k) dst[k] = v[k] * inv;
}

// ---------------------------------------------------------------------------
// Kernel 3 (fused): per (batch, 16-row block):
//   sim rows via V_WMMA_F32_16X16X4_F32  ->  LDS (16 x 1024 f32 = 64KB)
//   per-row top-32 (wave per row)        ->  softmax (lane k = element k)
//   broadcast gather-weighted sum from featT -> coalesced store via LDS
// 512 threads = 16 waves (wave32).
// ---------------------------------------------------------------------------
__global__ __launch_bounds__(512) void sim_topk_gather(
    const float* __restrict__ posn, const float* __restrict__ featT,
    float* __restrict__ out) {
  __shared__ float lds[16 * 1024];   // 64KB: sim rows, later reused as outbuf

  const int b    = blockIdx.y;
  const int r0   = blockIdx.x * 16;
  const int tid  = threadIdx.x;
  const int lane = tid & 31;
  const int w    = tid >> 5;          // wave id 0..15

  const float* pb = posn + (size_t)b * NPTS * POSD;

  // ----- A fragments: rows r0..r0+15, all 64 k, reused across column tiles.
  // f32 16x4 A layout: lanes 0-15 hold (K=4kk+0, 4kk+1), lanes 16-31 (+2,+3).
  const int mrow  = lane & 15;
  const int khalf = (lane >> 4) * 2;
  const float* arow = pb + (size_t)(r0 + mrow) * POSD;
  v2f afrag[16];
#pragma unroll
  for (int kk = 0; kk < 16; ++kk) {
    afrag[kk].x = arow[kk * 4 + khalf];
    afrag[kk].y = arow[kk * 4 + khalf + 1];
  }

  // ----- sim tiles: wave w computes column tiles {w, w+16, w+32, w+48}
  for (int t = 0; t < 4; ++t) {
    const int n0 = (w + t * 16) * 16;
    const float* brow = pb + (size_t)(n0 + mrow) * POSD;  // B[k][n] = posn[n][k]
    v8f acc = {};
#pragma unroll
    for (int kk = 0; kk < 16; ++kk) {
      v2f bfrag;
      bfrag.x = brow[kk * 4 + khalf];
      bfrag.y = brow[kk * 4 + khalf + 1];
      acc = __builtin_amdgcn_wmma_f32_16x16x4_f32(
          false, afrag[kk], false, bfrag, (short)0, acc, false, false);
    }
    // C layout: VGPR v -> (M = v + 8*(lane>=16), N = lane&15)
#pragma unroll
    for (int v = 0; v < 8; ++v) {
      int m = v + (lane >> 4) * 8;
      lds[m * 1024 + n0 + mrow] = acc[v];
    }
  }
  __syncthreads();

  // ----- top-32 of row w (wave w). Lane owns cols {j*32+lane}, j=0..31.
  float vals[32];
#pragma unroll
  for (int j = 0; j < 32; ++j) vals[j] = lds[w * 1024 + j * 32 + lane];

  unsigned taken = 0u;
  float myval = 0.0f;
  int   myidx = 0;
  for (int it = 0; it < KSEL; ++it) {
    float bestv = -__builtin_inff();
    int   bestj = -1;
#pragma unroll
    for (int j = 0; j < 32; ++j) {
      bool ok = ((taken >> j) & 1u) == 0u;
      if (ok && vals[j] > bestv) { bestv = vals[j]; bestj = j; }  // ties -> lowest col
    }
    int besti = (bestj < 0) ? 0x7fffffff : (bestj * 32 + lane);
    // wave reduce: max value, tie-break lowest index (matches jax top_k)
#pragma unroll
    for (int s = 16; s > 0; s >>= 1) {
      float ov = __shfl_xor(bestv, s, 32);
      int   oi = __shfl_xor(besti, s, 32);
      if (ov > bestv || (ov == bestv && oi < besti)) { bestv = ov; besti = oi; }
    }
    if (lane == it) { myval = bestv; myidx = besti; }
    if (besti < NPTS && (besti & 31) == lane) taken |= 1u << (besti >> 5);
  }
  __syncthreads();   // sim buffer dead; lds reused as output staging below

  // ----- softmax across the wave (lane k holds selected element k; lane 0 = max)
  float mx = __shfl(myval, 0, 32);
  float e  = __expf(myval - mx);
  float s  = e;
#pragma unroll
  for (int sft = 16; sft > 0; sft >>= 1) s += __shfl_xor(s, sft, 32);
  const float attn = e / s;

  // ----- gather weighted sum: lane covers channels {j*32+lane}, j=0..7
  const float* fb = featT + (size_t)b * NPTS * DIMF;
  float acc8[8] = {0.f, 0.f, 0.f, 0.f, 0.f, 0.f, 0.f, 0.f};
  for (int k = 0; k < KSEL; ++k) {
    float a   = __shfl(attn, k, 32);
    int   idx = __shfl(myidx, k, 32);
    const float* frow = fb + (size_t)idx * DIMF;   // contiguous 1KB row
#pragma unroll
    for (int j = 0; j < 8; ++j) acc8[j] += a * frow[j * 32 + lane];
  }

  // stage (row, channel) tile in LDS with pitch 257 (bank-conflict free)
#pragma unroll
  for (int j = 0; j < 8; ++j) lds[w * 257 + j * 32 + lane] = acc8[j];
  __syncthreads();

  // ----- coalesced transposed write: out[b][c][r0+rr], rr fastest
  float* ob = out + (size_t)b * DIMF * NPTS + r0;
  for (int i = tid; i < DIMF * 16; i += 512) {
    int rr = i & 15;
    int c  = i >> 4;
    ob[(size_t)c * NPTS + rr] = lds[rr * 257 + c];
  }
}

// ---------------------------------------------------------------------------
extern "C" void kernel_launch(void* const* d_in, const int* in_sizes, int n_in,
                              void* d_out, int out_size, void* d_ws, size_t ws_size,
                              hipStream_t stream) {
  (void)in_sizes; (void)n_in; (void)out_size; (void)ws_size;
  const float* in  = (const float*)d_in[0];
  float*       out = (float*)d_out;

  // workspace: featT (16MB) | posn (4MB)  -- both L2-resident (192MB L2)
  float* featT = (float*)d_ws;
  float* posn  = featT + (size_t)BATCH * NPTS * DIMF;

  feat_transpose<<<dim3(NPTS / 32, DIMF / 32, BATCH), dim3(32, 8), 0, stream>>>(in, featT);
  pos_normalize <<<dim3(NPTS / 256, BATCH), 256, 0, stream>>>(in, posn);
  sim_topk_gather<<<dim3(NPTS / 16, BATCH), 512, 0, stream>>>(posn, featT, out);
}